// VanillaGCNEncoder_80745385165161
// MI455X (gfx1250) — compile-verified
//
#include <hip/hip_runtime.h>
#include <hip/hip_bf16.h>

// ---------------------------------------------------------------------------
// VanillaGCNEncoder on MI455X (gfx1250, wave32).
// B=8, L=2048, K=30, D=128, NL=3.  fp32 end-to-end; GEMM via
// v_wmma_f32_16x16x4_f32 (exact fp32 matrix op on CDNA5).
// ---------------------------------------------------------------------------

#define GB   8
#define GL   2048
#define GK   30
#define GD   128
#define GNL  3

typedef __attribute__((ext_vector_type(2))) float v2f;
typedef __attribute__((ext_vector_type(8))) float v8f;

#define WAVES_PER_BLOCK 4
#define ROWS_PER_WAVE   16

// ---------------------------------------------------------------------------
// Transpose all NL weight matrices: Wt[layer][n][k] = W[layer][k][n]
// so WMMA B-fragments (adjacent K for fixed N) are contiguous 8-byte loads.
// ---------------------------------------------------------------------------
__global__ void gcn_transpose_w(const float* __restrict__ W,
                                float* __restrict__ Wt) {
    int i = blockIdx.x * blockDim.x + threadIdx.x;   // 0 .. NL*D*D-1
    if (i >= GNL * GD * GD) return;
    int layer = i / (GD * GD);
    int rem   = i % (GD * GD);
    int k     = rem / GD;       // input dim (row of W)
    int n     = rem % GD;       // output dim (col of W)
    Wt[layer * GD * GD + n * GD + k] = W[layer * GD * GD + k * GD + n];
}

// ---------------------------------------------------------------------------
// One GCN layer: gather-mean + residual + GEMM(WMMA fp32) + bias + ReLU +
// LayerNorm + mask.  One wave per 16-row tile; 4 waves per block.
// ---------------------------------------------------------------------------
__global__ __launch_bounds__(WAVES_PER_BLOCK * 32)
void gcn_layer(const float* __restrict__ h_in,        // [B, L, D]
               const long long* __restrict__ edge_idx, // [B, L, K] int64
               const float* __restrict__ mask,        // [B, L]
               const float* __restrict__ Wt,          // [D, D] transposed: Wt[n][k]
               const float* __restrict__ bias,        // [D]
               const float* __restrict__ gamma,       // [D]
               const float* __restrict__ beta,        // [D]
               float* __restrict__ h_out) {           // [B, L, D]
    // 4 waves * 16 rows * 128 cols * 4B = 32 KB
    __shared__ float xs[WAVES_PER_BLOCK][ROWS_PER_WAVE][GD];
    __shared__ float stats[WAVES_PER_BLOCK][ROWS_PER_WAVE][2]; // mu, rstd

    const int wave = threadIdx.x >> 5;
    const int lane = threadIdx.x & 31;

    const int tile = blockIdx.x * WAVES_PER_BLOCK + wave;  // global 16-row tile
    const int row0 = tile * ROWS_PER_WAVE;                 // flat b*L + l
    const int b    = row0 / GL;
    const int l0   = row0 % GL;                            // 16 | 2048 -> no batch crossing

    const float*     hb = h_in + (size_t)b * GL * GD;
    const long long* ib = edge_idx + ((size_t)b * GL + l0) * GK;

    // ---- gather + mean + residual: each lane owns 4 contiguous columns ----
    const int c0 = lane * 4;
    const float inv_k = 1.0f / (float)GK;
    for (int r = 0; r < ROWS_PER_WAVE; ++r) {
        const long long* ir = ib + r * GK;
        float4 acc = make_float4(0.f, 0.f, 0.f, 0.f);
        #pragma unroll 5
        for (int k = 0; k < GK; ++k) {
            const int nb = (int)ir[k];
            const float4 v = *(const float4*)(hb + (size_t)nb * GD + c0);
            acc.x += v.x; acc.y += v.y; acc.z += v.z; acc.w += v.w;
        }
        const float4 self = *(const float4*)(hb + (size_t)(l0 + r) * GD + c0);
        xs[wave][r][c0 + 0] = self.x + acc.x * inv_k;
        xs[wave][r][c0 + 1] = self.y + acc.y * inv_k;
        xs[wave][r][c0 + 2] = self.z + acc.z * inv_k;
        xs[wave][r][c0 + 3] = self.w + acc.w * inv_k;
    }
    __syncthreads();

    // ---- GEMM: Y[16x128] = X[16x128] * W[128x128] via WMMA f32 16x16x4 ----
    // A layout (ISA 7.12.2): lanes 0-15 hold M=lane, K={k0,k0+1};
    //                        lanes 16-31 hold M=lane-16, K={k0+2,k0+3}.
    // C/D layout: VGPR v -> row v (lanes 0-15) / row v+8 (lanes 16-31), N=lane%16.
    const int g = lane >> 4;        // K-pair group (0 or 1)
    const int m = lane & 15;        // M for A, N-within-tile for B/C/D

    v8f c[8];
    #pragma unroll
    for (int n = 0; n < 8; ++n) c[n] = (v8f)(0.0f);

    for (int k0 = 0; k0 < GD; k0 += 4) {
        const int kk = k0 + 2 * g;
        const float2 af = *(const float2*)&xs[wave][m][kk];
        v2f av; av.x = af.x; av.y = af.y;
        #pragma unroll
        for (int n = 0; n < 8; ++n) {
            // B frag: W[kk..kk+1][n*16+m] = Wt[n*16+m][kk..kk+1] (contiguous)
            const float2 bf = *(const float2*)(Wt + (size_t)(n * 16 + m) * GD + kk);
            v2f bv; bv.x = bf.x; bv.y = bf.y;
            c[n] = __builtin_amdgcn_wmma_f32_16x16x4_f32(
                false, av, false, bv, (short)0, c[n], false, false);
        }
    }
    __syncthreads();   // all A-frag reads done before xs is overwritten

    // ---- bias + ReLU, write Y back into the LDS tile ----
    #pragma unroll
    for (int n = 0; n < 8; ++n) {
        const int col = n * 16 + m;
        const float bb = bias[col];
        #pragma unroll
        for (int v = 0; v < 8; ++v) {
            const int row = v + 8 * g;
            float y = c[n][v] + bb;
            xs[wave][row][col] = y > 0.0f ? y : 0.0f;
        }
    }
    __syncthreads();

    // ---- per-row LayerNorm statistics (lanes 0-15, one row each) ----
    if (lane < ROWS_PER_WAVE) {
        float s = 0.0f, s2 = 0.0f;
        #pragma unroll 8
        for (int cidx = 0; cidx < GD; ++cidx) {
            const float y = xs[wave][lane][cidx];
            s += y; s2 += y * y;
        }
        const float mu  = s * (1.0f / GD);
        const float var = s2 * (1.0f / GD) - mu * mu;
        stats[wave][lane][0] = mu;
        stats[wave][lane][1] = rsqrtf(var + 1e-5f);
    }
    __syncthreads();

    // ---- normalize, scale, shift, mask, coalesced float4 store ----
    const float* mb = mask + (size_t)b * GL + l0;
    const float4 g4 = *(const float4*)(gamma + c0);
    const float4 b4 = *(const float4*)(beta + c0);
    for (int r = 0; r < ROWS_PER_WAVE; ++r) {
        const float mu = stats[wave][r][0];
        const float rs = stats[wave][r][1];
        const float mk = mb[r];
        const float4 y = *(const float4*)&xs[wave][r][c0];
        float4 o;
        o.x = ((y.x - mu) * rs * g4.x + b4.x) * mk;
        o.y = ((y.y - mu) * rs * g4.y + b4.y) * mk;
        o.z = ((y.z - mu) * rs * g4.z + b4.z) * mk;
        o.w = ((y.w - mu) * rs * g4.w + b4.w) * mk;
        *(float4*)(h_out + ((size_t)b * GL + l0 + r) * GD + c0) = o;
    }
}

// ---------------------------------------------------------------------------
extern "C" void kernel_launch(void* const* d_in, const int* in_sizes, int n_in,
                              void* d_out, int out_size, void* d_ws, size_t ws_size,
                              hipStream_t stream) {
    const float*     h_nodes = (const float*)d_in[0];
    // d_in[1] = h_edges: intentionally ignored (reference discards it)
    const long long* eidx    = (const long long*)d_in[2];   // int64
    const float*     mask    = (const float*)d_in[3];
    const float*     W       = (const float*)d_in[4];       // [NL, D, D]
    const float*     bias    = (const float*)d_in[5];       // [NL, D]
    const float*     gamma   = (const float*)d_in[6];       // [NL, D]
    const float*     beta    = (const float*)d_in[7];       // [NL, D]
    float*           out     = (float*)d_out;

    float* ws   = (float*)d_ws;
    float* Wt   = ws;                                   // NL*D*D floats (192 KB)
    float* buf0 = Wt + (size_t)GNL * GD * GD;           // B*L*D floats (8 MB)
    float* buf1 = buf0 + (size_t)GB * GL * GD;          // B*L*D floats (8 MB)

    // Transpose weights once (all layers).
    {
        const int total = GNL * GD * GD;
        gcn_transpose_w<<<(total + 255) / 256, 256, 0, stream>>>(W, Wt);
    }

    const int blocks = (GB * GL) / (ROWS_PER_WAVE * WAVES_PER_BLOCK);  // 256
    const int tpb    = WAVES_PER_BLOCK * 32;                           // 128

    // Layer 0: h_nodes -> buf0
    gcn_layer<<<blocks, tpb, 0, stream>>>(h_nodes, eidx, mask,
                                          Wt + 0 * GD * GD, bias + 0 * GD,
                                          gamma + 0 * GD, beta + 0 * GD, buf0);
    // Layer 1: buf0 -> buf1
    gcn_layer<<<blocks, tpb, 0, stream>>>(buf0, eidx, mask,
                                          Wt + 1 * GD * GD, bias + 1 * GD,
                                          gamma + 1 * GD, beta + 1 * GD, buf1);
    // Layer 2: buf1 -> d_out
    gcn_layer<<<blocks, tpb, 0, stream>>>(buf1, eidx, mask,
                                          Wt + 2 * GD * GD, bias + 2 * GD,
                                          gamma + 2 * GD, beta + 2 * GD, out);
}